// HeteroGAT_TATEncoder_47828755808349
// MI455X (gfx1250) — compile-verified
//
#include <hip/hip_runtime.h>
#include <cmath>

#define N_TX   100000
#define N_ADDR 150000
#define NE     1500000
#define F_TX   165
#define F_ADDR 64
#define HID    32
#define NH     2
#define HO     64
#define EMB    64

// Max K over all GEMMs is 165 -> rounded to multiple of 4 = 168 -> 84 K-pairs.
#define KPAIR_MAX 84

typedef float v2f __attribute__((ext_vector_type(2)));
typedef float v8f __attribute__((ext_vector_type(8)));

// ---------------------------------------------------------------------------
// WMMA fp32 GEMM:  C[M,N] = A[M,K] @ B[K,N] (+ bias[N])
// One wave computes one 16x16 tile of C using V_WMMA_F32_16X16X4_F32.
// M is always a multiple of 16 here (100000, 150000). N in {32, 64}.
// B is staged once per block into LDS as zero-padded (k, k+1) pairs so the
// hot loop is: 1 global A read + 1 ds_load_b64 + 1 v_wmma, no guards.
// Fragment layouts per CDNA5 ISA 7.12.2:
//   A (16x4): lane L -> row m = L%16, holds K = {kk, kk+1}, kk = 2*(L/16)
//   B (4x16): lane L -> col  = L%16, holds K = {kk, kk+1}
//   C:        VGPR r -> row r + 8*(L/16), col = L%16
// ---------------------------------------------------------------------------
__global__ __launch_bounds__(128)
void gemm_wmma_f32(const float* __restrict__ A, const float* __restrict__ B,
                   const float* __restrict__ bias, float* __restrict__ C,
                   int M, int K, int N) {
  __shared__ v2f bs[KPAIR_MAX * 64];     // 43008 B << 320 KB WGP LDS

  const int tid  = threadIdx.x;
  const int wave = tid >> 5;
  const int lane = tid & 31;
  const int row0 = blockIdx.x << 4;
  const int col0 = (blockIdx.y * 4 + wave) << 4;

  // ---- cooperative stage of B into LDS as zero-padded K-pairs -------------
  const int Kr    = (K + 3) & ~3;        // K rounded up to multiple of 4
  const int pairs = Kr >> 1;
  for (int i = tid; i < pairs * N; i += 128) {
    const int kp = i / N;
    const int n  = i - kp * N;
    const int k  = kp << 1;
    v2f w;
    w.x = (k     < K) ? B[(size_t)k * N + n]       : 0.0f;
    w.y = (k + 1 < K) ? B[(size_t)(k + 1) * N + n] : 0.0f;
    bs[kp * N + n] = w;
  }
  __syncthreads();

  if (col0 >= N) return;                 // wave-uniform exit after the barrier

  const int m  = lane & 15;              // row within A tile / col within B tile
  const int kh = lane >> 4;              // 0 or 1 -> K-pair select
  const int kk = kh << 1;                // 0 or 2

  v8f c = {};
  const float* arow = A + (size_t)(row0 + m) * K + kk;       // A[row, kk + k0]
  const v2f*   brow = bs + kh * N + (col0 + m);              // pair (k0/2 + kh)

  const int K4 = K & ~3;
  int k0 = 0;
  for (; k0 < K4; k0 += 4) {             // unguarded hot loop
    v2f av;
    av.x = arow[k0];
    av.y = arow[k0 + 1];
    const v2f bv = brow[(k0 >> 1) * N];
    c = __builtin_amdgcn_wmma_f32_16x16x4_f32(false, av, false, bv,
                                              (short)0, c, false, false);
  }
  if (k0 < K) {                          // single guarded tail (only K=165)
    const int ka = k0 + kk;
    const float a0 = A[(size_t)(row0 + m) * K + min(ka,     K - 1)];
    const float a1 = A[(size_t)(row0 + m) * K + min(ka + 1, K - 1)];
    v2f av;
    av.x = (ka     < K) ? a0 : 0.0f;     // v_cndmask, no branch
    av.y = (ka + 1 < K) ? a1 : 0.0f;
    const v2f bv = brow[(k0 >> 1) * N];  // LDS pairs are zero-padded to Kr
    c = __builtin_amdgcn_wmma_f32_16x16x4_f32(false, av, false, bv,
                                              (short)0, c, false, false);
  }

  const int col   = col0 + m;
  const int rbase = row0 + (kh << 3);
  const float bb  = bias ? bias[col] : 0.0f;
#pragma unroll
  for (int r = 0; r < 8; ++r)
    C[(size_t)(rbase + r) * N + col] = c[r] + bb;
}

// ---------------------------------------------------------------------------
// al[n,h] = dot(h[n, h*32 : h*32+32], a[h, :])
// ---------------------------------------------------------------------------
__global__ void node_alpha(const float* __restrict__ h, const float* __restrict__ a,
                           float* __restrict__ al, int Nn) {
  const int i = blockIdx.x * blockDim.x + threadIdx.x;
  if (i >= Nn * NH) return;
  const int n = i >> 1, hh = i & 1;
  const float* hp = h + (size_t)n * HO + hh * HID;
  const float* ap = a + hh * HID;
  float s = 0.0f;
#pragma unroll
  for (int c = 0; c < HID; ++c) s += hp[c] * ap[c];
  al[i] = s;
}

__global__ void fill_f32(float* __restrict__ p, float v, int n) {
  const int i = blockIdx.x * blockDim.x + threadIdx.x;
  if (i < n) p[i] = v;
}

// Two-sided float atomic max via integer atomics (IEEE ordering trick).
__device__ __forceinline__ void atomicMaxFloat(float* addr, float val) {
  if (val >= 0.0f)
    atomicMax((int*)addr, __float_as_int(val));
  else
    atomicMin((unsigned int*)addr, (unsigned int)__float_as_int(val));
}

// Pass 1: per-edge leaky-relu logit; segment max into mbuf[dst].
__global__ void edge_logit_max(const int* __restrict__ src, const int* __restrict__ dst,
                               const float* __restrict__ alS, const float* __restrict__ alD,
                               float* __restrict__ elog, float* __restrict__ mbuf, int E) {
  const int i = blockIdx.x * blockDim.x + threadIdx.x;
  if (i >= E * NH) return;
  const int e = i >> 1, h = i & 1;
  float l = alS[src[e] * NH + h] + alD[dst[e] * NH + h];
  l = (l > 0.0f) ? l : 0.2f * l;          // leaky_relu, slope 0.2
  elog[i] = l;
  atomicMaxFloat(&mbuf[dst[e] * NH + h], l);
}

// Pass 2: exp(logit - max); segment sum into sbuf[dst]; elog rewritten in place.
__global__ void edge_exp_sum(const int* __restrict__ dst, float* __restrict__ elog,
                             const float* __restrict__ mbuf, float* __restrict__ sbuf, int E) {
  const int i = blockIdx.x * blockDim.x + threadIdx.x;
  if (i >= E * NH) return;
  const int e = i >> 1, h = i & 1;
  const int d = dst[e] * NH + h;
  const float ee = __expf(elog[i] - mbuf[d]);
  elog[i] = ee;
  atomicAdd(&sbuf[d], ee);
}

// Pass 3: 64 threads per edge; coalesced gather of hs[src] and coalesced
// atomicAdd scatter into acc[dst] (acc stays L2-resident: <= 38 MB << 192 MB).
__global__ void edge_message(const int* __restrict__ src, const int* __restrict__ dst,
                             const float* __restrict__ hs, const float* __restrict__ elog,
                             const float* __restrict__ sbuf, float* __restrict__ acc, int E) {
  const int i = blockIdx.x * blockDim.x + threadIdx.x;   // E*64 = 96M < 2^31
  const int e = i >> 6;
  if (e >= E) return;
  const int t = i & 63;
  const int h = t >> 5;
  const int sn = src[e], dn = dst[e];
  const float alpha = elog[e * NH + h] / (sbuf[dn * NH + h] + 1e-16f);
  atomicAdd(&acc[(size_t)dn * HO + t], hs[(size_t)sn * HO + t] * alpha);
}

// Fused: x = acc + b;  y = LN(x)*g + be;  (+ residual);  ELU.  One wave/node.
__global__ void post_ln_elu(float* __restrict__ acc, const float* __restrict__ bias,
                            const float* __restrict__ g, const float* __restrict__ be,
                            const float* __restrict__ resid, int Nn) {
  const int node = blockIdx.x * (blockDim.x >> 5) + (threadIdx.x >> 5);
  const int lane = threadIdx.x & 31;
  if (node >= Nn) return;
  const size_t base = (size_t)node * HO;
  float x0 = acc[base + lane]      + bias[lane];
  float x1 = acc[base + lane + 32] + bias[lane + 32];
  float s = x0 + x1;
#pragma unroll
  for (int o = 16; o > 0; o >>= 1) s += __shfl_xor(s, o, 32);
  const float mu = s * (1.0f / 64.0f);
  const float d0 = x0 - mu, d1 = x1 - mu;
  float v = d0 * d0 + d1 * d1;
#pragma unroll
  for (int o = 16; o > 0; o >>= 1) v += __shfl_xor(v, o, 32);
  const float rinv = rsqrtf(v * (1.0f / 64.0f) + 1e-5f);
  float y0 = d0 * rinv * g[lane]      + be[lane];
  float y1 = d1 * rinv * g[lane + 32] + be[lane + 32];
  if (resid) { y0 += resid[base + lane]; y1 += resid[base + lane + 32]; }
  y0 = (y0 > 0.0f) ? y0 : __expf(y0) - 1.0f;   // ELU(alpha=1)
  y1 = (y1 > 0.0f) ? y1 : __expf(y1) - 1.0f;
  acc[base + lane]      = y0;
  acc[base + lane + 32] = y1;
}

// ---------------------------------------------------------------------------
extern "C" void kernel_launch(void* const* d_in, const int* in_sizes, int n_in,
                              void* d_out, int out_size, void* d_ws, size_t ws_size,
                              hipStream_t stream) {
  const float* x_tx    = (const float*)d_in[0];
  const float* x_addr  = (const float*)d_in[1];
  const float* Wp_tx   = (const float*)d_in[2];
  const float* bp_tx   = (const float*)d_in[3];
  const float* Wp_addr = (const float*)d_in[4];
  const float* bp_addr = (const float*)d_in[5];
  const float* g_tx    = (const float*)d_in[22];
  const float* be_tx   = (const float*)d_in[23];
  const float* g_addr  = (const float*)d_in[24];
  const float* be_addr = (const float*)d_in[25];
  const float* Wo      = (const float*)d_in[26];
  const float* bo      = (const float*)d_in[27];
  const int* e_src_ta  = (const int*)d_in[28];
  const int* e_dst_ta  = (const int*)d_in[29];
  const int* e_src_at  = (const int*)d_in[30];
  const int* e_dst_at  = (const int*)d_in[31];

  // ---- workspace carve-out -------------------------------------------------
  float* ws = (float*)d_ws;
  size_t off = 0;
  auto take = [&](size_t n) { float* p = ws + off; off += (n + 63) & ~(size_t)63; return p; };
  float* txA  = take((size_t)N_TX   * HO);
  float* txB  = take((size_t)N_TX   * HO);
  float* adA  = take((size_t)N_ADDR * HO);
  float* adB  = take((size_t)N_ADDR * HO);
  float* hs   = take((size_t)N_ADDR * HO);   // src-side projection (max count)
  float* hd   = take((size_t)N_ADDR * HO);   // dst-side projection (max count)
  float* alS  = take((size_t)N_ADDR * NH);
  float* alD  = take((size_t)N_ADDR * NH);
  float* mbuf = take((size_t)N_ADDR * NH);
  float* sbuf = take((size_t)N_ADDR * NH);
  float* elog = take((size_t)NE * NH);

  auto cdiv = [](int a, int b) { return (a + b - 1) / b; };
  auto gemm = [&](const float* A, const float* B, const float* bias, float* C,
                  int M, int K, int N) {
    dim3 grid(M / 16, cdiv(N, 64));
    gemm_wmma_f32<<<grid, 128, 0, stream>>>(A, B, bias, C, M, K, N);
  };
  auto fill = [&](float* p, float v, int n) {
    fill_f32<<<cdiv(n, 256), 256, 0, stream>>>(p, v, n);
  };

  // ---- input projections ---------------------------------------------------
  gemm(x_tx,   Wp_tx,   bp_tx,   txA, N_TX,   F_TX,   HID);   // [100k,32]
  gemm(x_addr, Wp_addr, bp_addr, adA, N_ADDR, F_ADDR, HID);   // [150k,32]

  float* txcur = txA; float* txnxt = txB;
  float* adcur = adA; float* adnxt = adB;
  const float NEG_INF = -INFINITY;

  // ---- two hetero-GAT layers ----------------------------------------------
  for (int l = 0; l < 2; ++l) {
    const int ind = l ? HO : HID;
    const float* Wta  = (const float*)d_in[l ? 14 : 6];
    const float* asta = (const float*)d_in[l ? 15 : 7];
    const float* adta = (const float*)d_in[l ? 16 : 8];
    const float* bta  = (const float*)d_in[l ? 17 : 9];
    const float* Wat  = (const float*)d_in[l ? 18 : 10];
    const float* asat = (const float*)d_in[l ? 19 : 11];
    const float* adat = (const float*)d_in[l ? 20 : 12];
    const float* bat  = (const float*)d_in[l ? 21 : 13];

    // relation tx->addr (produces adnxt)
    gemm(txcur, Wta, nullptr, hs, N_TX,   ind, HO);
    gemm(adcur, Wta, nullptr, hd, N_ADDR, ind, HO);
    node_alpha<<<cdiv(N_TX * NH, 256),   256, 0, stream>>>(hs, asta, alS, N_TX);
    node_alpha<<<cdiv(N_ADDR * NH, 256), 256, 0, stream>>>(hd, adta, alD, N_ADDR);
    fill(mbuf, NEG_INF, N_ADDR * NH);
    fill(sbuf, 0.0f,    N_ADDR * NH);
    fill(adnxt, 0.0f,   N_ADDR * HO);
    edge_logit_max<<<cdiv(NE * NH, 256), 256, 0, stream>>>(e_src_ta, e_dst_ta, alS, alD, elog, mbuf, NE);
    edge_exp_sum  <<<cdiv(NE * NH, 256), 256, 0, stream>>>(e_dst_ta, elog, mbuf, sbuf, NE);
    edge_message  <<<cdiv(NE * 64, 256), 256, 0, stream>>>(e_src_ta, e_dst_ta, hs, elog, sbuf, adnxt, NE);

    // relation addr->tx (produces txnxt)
    gemm(adcur, Wat, nullptr, hs, N_ADDR, ind, HO);
    gemm(txcur, Wat, nullptr, hd, N_TX,   ind, HO);
    node_alpha<<<cdiv(N_ADDR * NH, 256), 256, 0, stream>>>(hs, asat, alS, N_ADDR);
    node_alpha<<<cdiv(N_TX * NH, 256),   256, 0, stream>>>(hd, adat, alD, N_TX);
    fill(mbuf, NEG_INF, N_TX * NH);
    fill(sbuf, 0.0f,    N_TX * NH);
    fill(txnxt, 0.0f,   N_TX * HO);
    edge_logit_max<<<cdiv(NE * NH, 256), 256, 0, stream>>>(e_src_at, e_dst_at, alS, alD, elog, mbuf, NE);
    edge_exp_sum  <<<cdiv(NE * NH, 256), 256, 0, stream>>>(e_dst_at, elog, mbuf, sbuf, NE);
    edge_message  <<<cdiv(NE * 64, 256), 256, 0, stream>>>(e_src_at, e_dst_at, hs, elog, sbuf, txnxt, NE);

    // bias + LayerNorm + residual(layer>=1) + ELU
    post_ln_elu<<<cdiv(N_TX, 8),   256, 0, stream>>>(txnxt, bat, g_tx, be_tx,
                                                     l ? txcur : nullptr, N_TX);
    post_ln_elu<<<cdiv(N_ADDR, 8), 256, 0, stream>>>(adnxt, bta, g_addr, be_addr,
                                                     l ? adcur : nullptr, N_ADDR);

    float* t;
    t = txcur; txcur = txnxt; txnxt = t;
    t = adcur; adcur = adnxt; adnxt = t;
  }

  // ---- output head ---------------------------------------------------------
  gemm(txcur, Wo, bo, (float*)d_out, N_TX, HO, EMB);
}